// ConsNet_58669253263513
// MI455X (gfx1250) — compile-verified
//
#include <hip/hip_runtime.h>
#include <hip/hip_bf16.h>

// Problem sizes (fixed by the reference)
#define Bdim 8
#define Ldim 64
#define Fdim 64
#define Rdim 2048
#define Mdim (Bdim * Fdim)   // 512 rows of the GEMM

typedef __attribute__((ext_vector_type(2))) float v2f;
typedef __attribute__((ext_vector_type(8))) float v8f;

// ---------------------------------------------------------------------------
// Kernel 1: arg{1,2}[b,f,r] = sum_l x[b,l,f,r] * w{1,2}[b,l]
// One block per (b,f). 256 threads, each owns 8 consecutive r (2x float4).
// Streams the 256 MB x tensor once -> HBM-bound, fully coalesced b128 loads.
// ---------------------------------------------------------------------------
__global__ __launch_bounds__(256)
void consnet_reduce(const float* __restrict__ x,
                    const float* __restrict__ w1,
                    const float* __restrict__ w2,
                    float* __restrict__ arg1,
                    float* __restrict__ arg2) {
    const int bf = blockIdx.x;          // 0..511
    const int b  = bf >> 6;
    const int f  = bf & 63;
    const int t  = threadIdx.x;         // 0..255, owns r = t*8 .. t*8+7

    const size_t base    = ((size_t)(b * (Ldim * Fdim) + f)) * (size_t)Rdim + (size_t)t * 8;
    const size_t lstride = (size_t)Fdim * (size_t)Rdim;   // 131072 floats per l

    float a1[8] = {0,0,0,0,0,0,0,0};
    float a2[8] = {0,0,0,0,0,0,0,0};

    for (int l = 0; l < Ldim; ++l) {
        const float* p = x + base + (size_t)l * lstride;
        // prefetch a few L-iterations ahead (speculative, OOB is dropped)
        __builtin_prefetch(p + 8 * lstride, 0, 1);
        const float s1 = w1[b * Ldim + l];   // uniform -> scalar load
        const float s2 = w2[b * Ldim + l];
        const float4 xa = *(const float4*)(p);
        const float4 xb = *(const float4*)(p + 4);
        a1[0] = fmaf(xa.x, s1, a1[0]); a1[1] = fmaf(xa.y, s1, a1[1]);
        a1[2] = fmaf(xa.z, s1, a1[2]); a1[3] = fmaf(xa.w, s1, a1[3]);
        a1[4] = fmaf(xb.x, s1, a1[4]); a1[5] = fmaf(xb.y, s1, a1[5]);
        a1[6] = fmaf(xb.z, s1, a1[6]); a1[7] = fmaf(xb.w, s1, a1[7]);
        a2[0] = fmaf(xa.x, s2, a2[0]); a2[1] = fmaf(xa.y, s2, a2[1]);
        a2[2] = fmaf(xa.z, s2, a2[2]); a2[3] = fmaf(xa.w, s2, a2[3]);
        a2[4] = fmaf(xb.x, s2, a2[4]); a2[5] = fmaf(xb.y, s2, a2[5]);
        a2[6] = fmaf(xb.z, s2, a2[6]); a2[7] = fmaf(xb.w, s2, a2[7]);
    }

    float* o1 = arg1 + (size_t)bf * Rdim + (size_t)t * 8;
    float* o2 = arg2 + (size_t)bf * Rdim + (size_t)t * 8;
    *(float4*)(o1)     = make_float4(a1[0], a1[1], a1[2], a1[3]);
    *(float4*)(o1 + 4) = make_float4(a1[4], a1[5], a1[6], a1[7]);
    *(float4*)(o2)     = make_float4(a2[0], a2[1], a2[2], a2[3]);
    *(float4*)(o2 + 4) = make_float4(a2[4], a2[5], a2[6], a2[7]);
}

// ---------------------------------------------------------------------------
// Kernel 2: out[m,s] = sum_r arg1[m,r]*cons_l[s,r] + arg2[m,r]*cons_r[s,r]
//                      + root_filler[m] * root_role[s]
// FP32 GEMM via V_WMMA_F32_16X16X4_F32, 2x2 register-blocked:
// one wave32 computes a 32(M) x 32(N) region = 4 independent 16x16 C tiles.
// Per k-step: 8 v2f fragment loads feed 8 WMMAs on 4 independent
// accumulators (1:1 load:wmma, no accumulator RAW between adjacent WMMAs,
// and A/B fragments each reused twice -> ~1 GB total cache traffic).
//
// Fragment layouts per CDNA5 ISA (CDNA5 ISA):
//   A 16x4 f32:  lane = half*16 + idx ; VGPR v holds A[mBase+idx][k + half*2 + v]
//   B 4x16 f32:  mirrored:            VGPR v holds B[k + half*2 + v][sBase+idx]
//     With einsum 'bfr,sr->bfs', B[k][s] = cons[s][k]; cons row-major in r=k,
//     so each lane's 2 elements are a contiguous float2: cons[sBase+idx][k+half*2..]
//   C 16x16 f32: VGPR v -> M = v + half*8, N = idx
// ---------------------------------------------------------------------------
__global__ __launch_bounds__(256)
void consnet_gemm(const float* __restrict__ arg1,
                  const float* __restrict__ arg2,
                  const float* __restrict__ cons_l,
                  const float* __restrict__ cons_r,
                  const float* __restrict__ root_filler,
                  const float* __restrict__ root_role,
                  float* __restrict__ out) {
    const int lane = threadIdx.x & 31;
    const int wave = threadIdx.x >> 5;
    const int tile = blockIdx.x * 8 + wave;      // 1024 32x32 regions total
    const int sTile = tile & ((Rdim / 32) - 1);  // 64 N-regions
    const int mTile = tile >> 6;                 // 16 M-regions
    const int mBase = mTile * 32;
    const int sBase = sTile * 32;

    const int half = lane >> 4;   // 0: lanes 0-15, 1: lanes 16-31
    const int idx  = lane & 15;

    // A row pointers: two 16-row M tiles x two matrices
    const float* a1p0 = arg1   + (size_t)(mBase + idx)      * Rdim + half * 2;
    const float* a1p1 = arg1   + (size_t)(mBase + 16 + idx) * Rdim + half * 2;
    const float* a2p0 = arg2   + (size_t)(mBase + idx)      * Rdim + half * 2;
    const float* a2p1 = arg2   + (size_t)(mBase + 16 + idx) * Rdim + half * 2;
    // B row pointers: two 16-col N tiles x two matrices
    const float* blp0 = cons_l + (size_t)(sBase + idx)      * Rdim + half * 2;
    const float* blp1 = cons_l + (size_t)(sBase + 16 + idx) * Rdim + half * 2;
    const float* brp0 = cons_r + (size_t)(sBase + idx)      * Rdim + half * 2;
    const float* brp1 = cons_r + (size_t)(sBase + 16 + idx) * Rdim + half * 2;

    v8f c00 = {0.f, 0.f, 0.f, 0.f, 0.f, 0.f, 0.f, 0.f};
    v8f c01 = c00, c10 = c00, c11 = c00;

    for (int k = 0; k < Rdim; k += 4) {
        const v2f fa1m0 = *(const v2f*)(a1p0 + k);
        const v2f fa1m1 = *(const v2f*)(a1p1 + k);
        const v2f fa2m0 = *(const v2f*)(a2p0 + k);
        const v2f fa2m1 = *(const v2f*)(a2p1 + k);
        const v2f fbl0  = *(const v2f*)(blp0 + k);
        const v2f fbl1  = *(const v2f*)(blp1 + k);
        const v2f fbr0  = *(const v2f*)(brp0 + k);
        const v2f fbr1  = *(const v2f*)(brp1 + k);
        // 8 WMMAs, round-robin over 4 independent accumulators
        c00 = __builtin_amdgcn_wmma_f32_16x16x4_f32(
                  false, fa1m0, false, fbl0, (short)0, c00, false, false);
        c01 = __builtin_amdgcn_wmma_f32_16x16x4_f32(
                  false, fa1m0, false, fbl1, (short)0, c01, false, false);
        c10 = __builtin_amdgcn_wmma_f32_16x16x4_f32(
                  false, fa1m1, false, fbl0, (short)0, c10, false, false);
        c11 = __builtin_amdgcn_wmma_f32_16x16x4_f32(
                  false, fa1m1, false, fbl1, (short)0, c11, false, false);
        c00 = __builtin_amdgcn_wmma_f32_16x16x4_f32(
                  false, fa2m0, false, fbr0, (short)0, c00, false, false);
        c01 = __builtin_amdgcn_wmma_f32_16x16x4_f32(
                  false, fa2m0, false, fbr1, (short)0, c01, false, false);
        c10 = __builtin_amdgcn_wmma_f32_16x16x4_f32(
                  false, fa2m1, false, fbr0, (short)0, c10, false, false);
        c11 = __builtin_amdgcn_wmma_f32_16x16x4_f32(
                  false, fa2m1, false, fbr1, (short)0, c11, false, false);
    }

    // Epilogue: + root_filler[b,f] * root_role[s]
    const int s0 = sBase + idx;
    const int s1 = sBase + 16 + idx;
    const float role0 = root_role[s0];
    const float role1 = root_role[s1];
#pragma unroll
    for (int v = 0; v < 8; ++v) {
        const int m0 = mBase + half * 8 + v;      // b = m/64, f = m%64
        const int m1 = m0 + 16;
        const float rf0 = root_filler[m0];        // root_filler is [B,F] flat = [m]
        const float rf1 = root_filler[m1];
        out[(size_t)m0 * Rdim + s0] = c00[v] + rf0 * role0;
        out[(size_t)m0 * Rdim + s1] = c01[v] + rf0 * role1;
        out[(size_t)m1 * Rdim + s0] = c10[v] + rf1 * role0;
        out[(size_t)m1 * Rdim + s1] = c11[v] + rf1 * role1;
    }
}

// ---------------------------------------------------------------------------
// Kernel 3: max over L of the two weight tensors. One wave32 per b.
// ---------------------------------------------------------------------------
__global__ __launch_bounds__(256)
void consnet_max(const float* __restrict__ w1,
                 const float* __restrict__ w2,
                 float* __restrict__ max1,
                 float* __restrict__ max2) {
    const int lane = threadIdx.x & 31;
    const int b    = threadIdx.x >> 5;           // 8 waves -> 8 batches
    float m1 = fmaxf(w1[b * Ldim + lane], w1[b * Ldim + lane + 32]);
    float m2 = fmaxf(w2[b * Ldim + lane], w2[b * Ldim + lane + 32]);
#pragma unroll
    for (int off = 16; off > 0; off >>= 1) {
        m1 = fmaxf(m1, __shfl_xor(m1, off, 32));
        m2 = fmaxf(m2, __shfl_xor(m2, off, 32));
    }
    if (lane == 0) {
        max1[b] = m1;
        max2[b] = m2;
    }
}

extern "C" void kernel_launch(void* const* d_in, const int* in_sizes, int n_in,
                              void* d_out, int out_size, void* d_ws, size_t ws_size,
                              hipStream_t stream) {
    const float* x      = (const float*)d_in[0];   // [B,L,F,R]
    const float* w1     = (const float*)d_in[1];   // [B,L]
    const float* w2     = (const float*)d_in[2];   // [B,L]
    const float* rootf  = (const float*)d_in[3];   // [B,F]
    const float* cons_l = (const float*)d_in[4];   // [R,R]
    const float* cons_r = (const float*)d_in[5];   // [R,R]
    const float* rootr  = (const float*)d_in[6];   // [R]

    float* out  = (float*)d_out;                         // [B,F,R] flat
    float* max1 = out + (size_t)Mdim * Rdim;             // [B]
    float* max2 = max1 + Bdim;                           // [B]

    float* arg1 = (float*)d_ws;                          // [M,R] = 4 MB
    float* arg2 = arg1 + (size_t)Mdim * Rdim;            // [M,R] = 4 MB

    consnet_reduce<<<Mdim, 256, 0, stream>>>(x, w1, w2, arg1, arg2);
    // 1024 waves = (512/32) * (2048/32) regions, 8 waves per block
    consnet_gemm<<<(Mdim / 32) * (Rdim / 32) / 8, 256, 0, stream>>>(
        arg1, arg2, cons_l, cons_r, rootf, rootr, out);
    consnet_max<<<1, 256, 0, stream>>>(w1, w2, max1, max2);
}